// GraphAttentionLayer_71708773974389
// MI455X (gfx1250) — compile-verified
//
#include <hip/hip_runtime.h>
#include <math.h>
#include <stdint.h>

typedef __attribute__((ext_vector_type(2))) float v2f;
typedef __attribute__((ext_vector_type(8))) float v8f;

#define B_      4
#define N_      2048
#define DIN_    128
#define H_      4
#define HD_     32
#define OUTD_   128       // H_*HD_
#define KSEL_   204       // int(0.1 * 2048)
#define KPAD_   208       // KSEL_ rounded up to multiple of 16
#define KCHUNKS (KPAD_/4) // 52
#define LEAKY_  0.2f
#define NEGINF_ -1.0e9f
#define ASTRIDE 132       // LDS A-tile row stride (x4 dwords -> 16B-aligned rows,
                          // 132 % 64 = 4 -> 16 lanes cover 16 distinct banks)

__device__ __forceinline__ float lrelu(float x) { return x > 0.0f ? x : LEAKY_ * x; }

// CDNA5 async global->LDS copy (ASYNCcnt path). VDST = LDS byte offset VGPR,
// VADDR = 64-bit global address. Flat pointers to LDS carry the LDS byte
// offset in their low 32 bits (ISA 10.2 aperture mapping).
__device__ __forceinline__ void async_copy_b128(uint32_t lds_off, const void* gaddr) {
    asm volatile("global_load_async_to_lds_b128 %0, %1, off"
                 :: "v"(lds_off), "v"(gaddr)
                 : "memory");
}
__device__ __forceinline__ void wait_async_all() {
    asm volatile("s_wait_asynccnt 0" ::: "memory");
}

// ---------------------------------------------------------------------------
// Kernel 0: pack W [H][Din][hd] into Wt[col][f] (col = h*32+d) so that WMMA
// B-operand fetches are contiguous float2 along K (=f).
// ---------------------------------------------------------------------------
__global__ __launch_bounds__(256) void k_pack_w(const float* __restrict__ W,
                                                float* __restrict__ Wt) {
    int t = blockIdx.x * 256 + threadIdx.x;     // 128*128 elements
    if (t >= OUTD_ * DIN_) return;
    int col = t >> 7;          // 0..127
    int f   = t & 127;         // 0..127
    int h   = col >> 5;
    int d   = col & 31;
    Wt[col * DIN_ + f] = W[h * (DIN_ * HD_) + f * HD_ + d];
}

// ---------------------------------------------------------------------------
// Kernel 1: Wh[b] = h[b] (2048x128) @ Wcat (128x128) via V_WMMA_F32_16X16X4_F32.
// grid = (N/16, B), 256 threads = 8 waves; each wave owns one 16-col tile;
// all waves share one 16-row A tile staged in LDS via async global->LDS b128.
// ---------------------------------------------------------------------------
__global__ __launch_bounds__(256) void k_gemm_wh(const float* __restrict__ h,
                                                 const float* __restrict__ Wt,
                                                 float* __restrict__ Wh) {
    __shared__ float ldsA[16 * ASTRIDE];
    const int b       = blockIdx.y;
    const int rowbase = blockIdx.x * 16;
    const int tid     = threadIdx.x;

    // async-stage A tile: 16 rows x 128 floats = 512 b128 chunks, 2 per thread
    const float* gA = h + ((size_t)(b * N_ + rowbase)) * DIN_;
    #pragma unroll
    for (int c = tid; c < 16 * 32; c += 256) {
        int r = c >> 5;            // row 0..15
        int q = c & 31;            // b128 chunk within row
        uint32_t lds_off = (uint32_t)(uintptr_t)&ldsA[r * ASTRIDE + q * 4];
        async_copy_b128(lds_off, gA + (size_t)r * DIN_ + q * 4);
    }
    wait_async_all();
    __syncthreads();

    const int lane   = tid & 31;
    const int wave   = tid >> 5;
    const int lane15 = lane & 15;
    const int khalf  = lane >> 4;        // 0 -> K{0,1}, 1 -> K{2,3}
    const int colbase = wave * 16;

    v8f acc = {};
    const float* bcol = Wt + (size_t)(colbase + lane15) * DIN_;
    #pragma unroll 4
    for (int kc = 0; kc < DIN_ / 4; ++kc) {
        int k0 = kc * 4 + 2 * khalf;
        v2f av = *(const v2f*)&ldsA[lane15 * ASTRIDE + k0];
        v2f bv = *(const v2f*)&bcol[k0];
        acc = __builtin_amdgcn_wmma_f32_16x16x4_f32(
            false, av, false, bv, (short)0, acc, false, false);
    }

    // C/D layout: VGPR v -> row v + 8*khalf, col = lane15
    float* gC = Wh + ((size_t)(b * N_ + rowbase)) * OUTD_ + colbase + lane15;
    #pragma unroll
    for (int v = 0; v < 8; ++v) {
        gC[(size_t)(v + 8 * khalf) * OUTD_] = acc[v];
    }
}

// ---------------------------------------------------------------------------
// Kernel 2: s_i / s_j per (b,h,n): dot(Wh row head-slice, a halves).
// ---------------------------------------------------------------------------
__global__ __launch_bounds__(256) void k_sij(const float* __restrict__ Wh,
                                             const float* __restrict__ a,
                                             float* __restrict__ s_i,
                                             float* __restrict__ s_j) {
    int t = blockIdx.x * 256 + threadIdx.x;     // B*H*N = 32768
    if (t >= B_ * H_ * N_) return;
    int n  = t & (N_ - 1);
    int bh = t >> 11;
    int hh = bh & (H_ - 1);
    int b  = bh >> 2;
    const float* row = Wh + ((size_t)(b * N_ + n)) * OUTD_ + hh * HD_;
    const float* av  = a + hh * (2 * HD_);
    float si = 0.0f, sj = 0.0f;
    #pragma unroll
    for (int d = 0; d < HD_; ++d) {
        float w = row[d];
        si = fmaf(w, av[d], si);
        sj = fmaf(w, av[HD_ + d], sj);
    }
    s_i[t] = si;
    s_j[t] = sj;
}

// ---------------------------------------------------------------------------
// Kernel 3: per (b,h) bitonic sort of 2048 (s_j, idx) pairs, descending by
// value with ascending index tie-break (matches lax.top_k), keep top 204.
// ---------------------------------------------------------------------------
__global__ __launch_bounds__(1024) void k_topk(const float* __restrict__ s_j,
                                               float* __restrict__ sval_sel,
                                               int* __restrict__ idx_sel) {
    __shared__ float v[N_];
    __shared__ int   id[N_];
    const int bh = blockIdx.x;
    const int t  = threadIdx.x;

    v[t]        = s_j[bh * N_ + t];        id[t]        = t;
    v[t + 1024] = s_j[bh * N_ + t + 1024]; id[t + 1024] = t + 1024;

    for (int k = 2; k <= N_; k <<= 1) {
        for (int j = k >> 1; j > 0; j >>= 1) {
            __syncthreads();
            #pragma unroll
            for (int base = 0; base < N_; base += 1024) {
                int i = base + t;
                int l = i ^ j;
                if (l > i) {
                    float vi = v[i], vl = v[l];
                    int   ii = id[i], il = id[l];
                    bool i_first = (vi > vl) || (vi == vl && ii < il);
                    bool desc    = ((i & k) == 0);
                    if (desc != i_first) {
                        v[i] = vl; v[l] = vi;
                        id[i] = il; id[l] = ii;
                    }
                }
            }
        }
    }
    __syncthreads();
    if (t < KSEL_) {
        sval_sel[bh * KSEL_ + t] = v[t];
        idx_sel[bh * KSEL_ + t]  = id[t];
    }
}

// ---------------------------------------------------------------------------
// Kernel 4: per-row softmax stats over the 204 selected entries.
// One wave per row; shfl_xor reductions (wave32).
// ---------------------------------------------------------------------------
__global__ __launch_bounds__(256) void k_stats(const float* __restrict__ s_i,
                                               const float* __restrict__ sval_sel,
                                               float* __restrict__ rowmax,
                                               float* __restrict__ invden) {
    const int wave = threadIdx.x >> 5;
    const int lane = threadIdx.x & 31;
    const int r    = blockIdx.x * 8 + wave;       // row in [0, B*H*N)
    const int bh   = r >> 11;
    const float si = s_i[r];
    const float* sv = sval_sel + bh * KSEL_;

    float m = -3.4e38f;
    for (int kk = lane; kk < KSEL_; kk += 32)
        m = fmaxf(m, lrelu(si + sv[kk]));
    #pragma unroll
    for (int off = 16; off > 0; off >>= 1)
        m = fmaxf(m, __shfl_xor(m, off, 32));

    float den = 0.0f;
    for (int kk = lane; kk < KSEL_; kk += 32)
        den += __expf(lrelu(si + sv[kk]) - m);
    #pragma unroll
    for (int off = 16; off > 0; off >>= 1)
        den += __shfl_xor(den, off, 32);

    if (lane == 0) {
        rowmax[r] = m;
        invden[r] = 1.0f / den;
    }
}

// ---------------------------------------------------------------------------
// Kernel 5: out tile = attn_sel (128 x 204) @ Wh_sel (204 x 32) per (b,h)
// with WMMA f32 16x16x4. Attention A-operands generated on the fly per lane;
// Wh_sel staged transposed in LDS (K-contiguous, stride 210). Pad K entries
// use sval = -1e9 so exp underflows to exactly 0 -> attn contribution 0.
// grid = (N/128, B*H), 256 threads = 8 waves (16 rows each, 2 col tiles).
// ---------------------------------------------------------------------------
__global__ __launch_bounds__(256) void k_out_gemm(const float* __restrict__ Wh,
                                                  const float* __restrict__ adj,
                                                  const float* __restrict__ s_i,
                                                  const float* __restrict__ sval_sel,
                                                  const int* __restrict__ idx_sel,
                                                  const float* __restrict__ rowmax,
                                                  const float* __restrict__ invden,
                                                  float* __restrict__ out) {
    __shared__ float sWhT[HD_ * 210];   // [d][kk], kk stride 210 (even, !=64k)
    __shared__ float sSval[KPAD_];
    __shared__ int   sIdx[KPAD_];

    const int bh = blockIdx.y;
    const int hh = bh & (H_ - 1);
    const int b  = bh >> 2;
    const int rowblock = blockIdx.x * 128;
    const int tid = threadIdx.x;

    // stage selected s_j values / indices / Wh rows (head slice, transposed)
    for (int kk = tid; kk < KPAD_; kk += 256) {
        int src = (kk < KSEL_) ? idx_sel[bh * KSEL_ + kk] : -1;
        sSval[kk] = (src >= 0) ? sval_sel[bh * KSEL_ + kk] : NEGINF_;
        sIdx[kk]  = (src >= 0) ? src : 0;
        const float* row = Wh + ((size_t)(b * N_ + (src >= 0 ? src : 0))) * OUTD_ + hh * HD_;
        #pragma unroll
        for (int d = 0; d < HD_; ++d)
            sWhT[d * 210 + kk] = (src >= 0) ? row[d] : 0.0f;
    }
    __syncthreads();

    const int lane   = tid & 31;
    const int wave   = tid >> 5;
    const int lane15 = lane & 15;
    const int khalf  = lane >> 4;
    const int rowbase = rowblock + wave * 16;
    const int i = rowbase + lane15;               // this lane's attention row
    const int gr = bh * N_ + i;

    const float si   = s_i[gr];
    const float m    = rowmax[gr];
    const float invd = invden[gr];
    const float* adjrow = adj + ((size_t)b * N_ + i) * N_;
    __builtin_prefetch(adjrow, 0, 1);             // warm GL2 for the gather loop

    v8f c0 = {}, c1 = {};
    for (int kc = 0; kc < KCHUNKS; ++kc) {
        int k0 = kc * 4 + 2 * khalf;
        // attention values (A operand): exp(lrelu(si+sj)-m) * invden * adj
        float a0 = __expf(lrelu(si + sSval[k0])     - m) * invd * adjrow[sIdx[k0]];
        float a1 = __expf(lrelu(si + sSval[k0 + 1]) - m) * invd * adjrow[sIdx[k0 + 1]];
        v2f av = {a0, a1};
        v2f b0 = *(const v2f*)&sWhT[lane15 * 210 + k0];         // cols 0..15
        v2f b1 = *(const v2f*)&sWhT[(16 + lane15) * 210 + k0];  // cols 16..31
        c0 = __builtin_amdgcn_wmma_f32_16x16x4_f32(
            false, av, false, b0, (short)0, c0, false, false);
        c1 = __builtin_amdgcn_wmma_f32_16x16x4_f32(
            false, av, false, b1, (short)0, c1, false, false);
    }

    // D layout: VGPR v -> row v + 8*khalf, col = lane15 (tile0) / 16+lane15 (tile1)
    #pragma unroll
    for (int v = 0; v < 8; ++v) {
        int r = rowbase + v + 8 * khalf;
        size_t o = ((size_t)(b * N_ + r)) * OUTD_ + hh * HD_;
        out[o + lane15]      = c0[v];
        out[o + 16 + lane15] = c1[v];
    }
}

// ---------------------------------------------------------------------------
extern "C" void kernel_launch(void* const* d_in, const int* in_sizes, int n_in,
                              void* d_out, int out_size, void* d_ws, size_t ws_size,
                              hipStream_t stream) {
    const float* h   = (const float*)d_in[0];   // [B,N,Din]
    const float* adj = (const float*)d_in[1];   // [B,N,N]
    const float* W   = (const float*)d_in[2];   // [H,Din,hd]
    const float* a   = (const float*)d_in[3];   // [H,2*hd]
    float* out = (float*)d_out;                 // [B,N,H*hd]

    // workspace layout (floats)
    float* ws = (float*)d_ws;
    float* Wt       = ws;                       // 128*128          = 16384
    float* Wh       = Wt + OUTD_ * DIN_;        // B*N*128          = 1048576
    float* s_i      = Wh + (size_t)B_ * N_ * OUTD_;      // 32768
    float* s_j      = s_i + B_ * H_ * N_;                // 32768
    float* sval_sel = s_j + B_ * H_ * N_;                // 16*204 (alloc KPAD_)
    int*   idx_sel  = (int*)(sval_sel + B_ * H_ * KPAD_);
    float* rowmax   = (float*)(idx_sel + B_ * H_ * KPAD_);
    float* invden   = rowmax + B_ * H_ * N_;

    k_pack_w<<<(OUTD_ * DIN_ + 255) / 256, 256, 0, stream>>>(W, Wt);
    k_gemm_wh<<<dim3(N_ / 16, B_), 256, 0, stream>>>(h, Wt, Wh);
    k_sij<<<(B_ * H_ * N_) / 256, 256, 0, stream>>>(Wh, a, s_i, s_j);
    k_topk<<<B_ * H_, 1024, 0, stream>>>(s_j, sval_sel, idx_sel);
    k_stats<<<(B_ * H_ * N_) / 8, 256, 0, stream>>>(s_i, sval_sel, rowmax, invden);
    k_out_gemm<<<dim3(N_ / 128, B_ * H_), 256, 0, stream>>>(
        Wh, adj, s_i, sval_sel, idx_sel, rowmax, invden, out);
}